// KGICLPromptEnhancer_59708635349479
// MI455X (gfx1250) — compile-verified
//
#include <hip/hip_runtime.h>
#include <math.h>
#include <stdint.h>

// Problem constants (reference: B=256, R=2048, H=64, E=20)
constexpr int Bn = 256;
constexpr int Rn = 2048;
constexpr int Hn = 64;
constexpr int En = 20;
constexpr float LN_EPS = 1e-5f;

typedef float v2f __attribute__((ext_vector_type(2)));
typedef float v8f __attribute__((ext_vector_type(8)));

// ---------------------------------------------------------------------------
// CDNA5 async global->LDS helpers (ASYNCcnt-tracked DMA path, no VGPR bounce)
// ---------------------------------------------------------------------------
__device__ __forceinline__ void async_copy_b128(void* lds_dst, const void* gsrc) {
  // GLOBAL_LOAD_ASYNC_TO_LDS_B128, GV mode: vdst = per-lane LDS byte address
  // (low 32 bits of generic pointer = LDS offset per flat-aperture rules).
  uint32_t lds_off = (uint32_t)(uintptr_t)lds_dst;
  uint64_t gaddr   = (uint64_t)(uintptr_t)gsrc;
  asm volatile("global_load_async_to_lds_b128 %0, %1, off"
               :: "v"(lds_off), "v"(gaddr)
               : "memory");
}

__device__ __forceinline__ void wait_async0() {
#if __has_builtin(__builtin_amdgcn_s_wait_asynccnt)
  __builtin_amdgcn_s_wait_asynccnt(0);
#else
  asm volatile("s_wait_asynccnt 0" ::: "memory");
#endif
}

// ---------------------------------------------------------------------------
// Kernel 1: per-sample sparse prompt rows.
// Only rows named in edge_type[b] differ from the "default" prompt row
// (LN(upd_b)).  Compute those <=20 rows per sample and stash them in d_ws.
// ---------------------------------------------------------------------------
__global__ __launch_bounds__(64) void prompt_sparse_kernel(
    const int* __restrict__ query_relations,
    const int* __restrict__ edge_type,
    const float* __restrict__ init_noise,
    const float* __restrict__ msg_W,
    const float* __restrict__ msg_b,
    const float* __restrict__ upd_W,
    const float* __restrict__ upd_b,
    const float* __restrict__ ln_g,
    const float* __restrict__ ln_b,
    int* __restrict__ special_ids,
    float* __restrict__ special_rows)
{
  __shared__ float sWeff[Hn * Hn];   // msg_W[:H] + msg_W[H:], 64x64
  __shared__ float sUpdW[Hn * Hn];
  __shared__ float sH[En][Hn];
  __shared__ float sAgg[En][Hn];
  __shared__ float sUpd[En][Hn];
  __shared__ float sMu[En], sRs[En];
  __shared__ int   sEdge[En], sOwner[En];

  const int j = threadIdx.x;          // 0..63 (column)
  const int b = blockIdx.x;           // sample
  const int qr = query_relations[b];

  if (j < En) sEdge[j] = edge_type[b * En + j];
  for (int k = 0; k < Hn; ++k) {
    sWeff[k * Hn + j] = msg_W[k * Hn + j] + msg_W[(k + Hn) * Hn + j];
    sUpdW[k * Hn + j] = upd_W[k * Hn + j];
  }
  __syncthreads();

  if (j < En) {
    int o = j;                        // owner = first slot with same type
    for (int e = 0; e < j; ++e)
      if (sEdge[e] == sEdge[j]) { o = e; break; }
    sOwner[j] = o;
  }
  // gather h rows (query row is all-ones, others init_noise*0.1); zero agg
  for (int e = 0; e < En; ++e) {
    int t = sEdge[e];
    float v = (t == qr) ? 1.0f
                        : init_noise[(((long)b * Rn + t) << 6) + j] * 0.1f;
    sH[e][j]   = v;
    sAgg[e][j] = 0.0f;
  }
  __syncthreads();

  // msg = relu(h @ Weff + msg_b), segment-sum into owner slot
  for (int e = 0; e < En; ++e) {
    float acc = msg_b[j];
    for (int k = 0; k < Hn; ++k) acc += sH[e][k] * sWeff[k * Hn + j];
    acc = fmaxf(acc, 0.0f);
    sAgg[sOwner[e]][j] += acc;        // thread j owns column j: no race
  }
  __syncthreads();

  // upd = agg @ upd_W + upd_b (owner slots only)
  for (int e = 0; e < En; ++e) {
    if (sOwner[e] == e) {
      float u = upd_b[j];
      for (int k = 0; k < Hn; ++k) u += sAgg[e][k] * sUpdW[k * Hn + j];
      sUpd[e][j] = u;
    }
  }
  __syncthreads();

  if (j < En && sOwner[j] == j) {     // LayerNorm stats per owner row
    float mu = 0.0f;
    for (int k = 0; k < Hn; ++k) mu += sUpd[j][k];
    mu *= (1.0f / Hn);
    float var = 0.0f;
    for (int k = 0; k < Hn; ++k) { float d = sUpd[j][k] - mu; var += d * d; }
    var *= (1.0f / Hn);
    sMu[j] = mu;
    sRs[j] = rsqrtf(var + LN_EPS);
  }
  __syncthreads();

  for (int e = 0; e < En; ++e) {
    bool own = (sOwner[e] == e);
    if (j == 0) special_ids[b * En + e] = own ? sEdge[e] : -1;
    if (own) {
      special_rows[(((long)b * En + e) << 6) + j] =
          (sUpd[e][j] - sMu[e]) * sRs[e] * ln_g[j] + ln_b[j];
    }
  }
}

// ---------------------------------------------------------------------------
// Kernel 2: fusion + gating over all B*R rows with V_WMMA_F32_16X16X4_F32.
// One block = 128 rows x 64 cols; 8 waves, each wave owns a 16-row strip.
// Staging of base/W1/W2 uses GLOBAL_LOAD_ASYNC_TO_LDS_B128 (ASYNCcnt DMA).
// ---------------------------------------------------------------------------
constexpr int ROWS = 128;
constexpr int XS   = 132;   // X row stride (pad 4 -> conflict-free A frags), 528B (16B-aligned rows)
constexpr int WS   = 72;    // W1/W2 row stride (pad 8 -> conflict-free B frags), 288B (16B-aligned rows)

__global__ __launch_bounds__(256) void fuse_gate_kernel(
    const float* __restrict__ base,
    const float* __restrict__ fus_W1, const float* __restrict__ fus_b1,
    const float* __restrict__ fus_W2, const float* __restrict__ fus_b2,
    const float* __restrict__ gate_W, const float* __restrict__ gate_b,
    const float* __restrict__ upd_b,  const float* __restrict__ ln_g,
    const float* __restrict__ ln_b,
    const int* __restrict__ special_ids,
    const float* __restrict__ special_rows,
    float* __restrict__ out)
{
  __shared__ float Xs[ROWS * XS];     // [128][132]: cols 0..63 base, 64..127 prompt (later relu-Y overlay)
  __shared__ float W1s[128 * WS];
  __shared__ float W2s[64 * WS];
  __shared__ float gWs[128];
  __shared__ float b1s[Hn], b2s[Hn], dps[Hn];
  __shared__ float gates[ROWS];
  __shared__ float stats[2];

  const int tid  = threadIdx.x;
  const int b    = blockIdx.x >> 4;
  const int row0 = (blockIdx.x & 15) * ROWS;
  const size_t gbase = ((size_t)b * Rn + row0) << 6;   // *64 floats

  // --- async stage: base rows -> Xs[:,0:64]  (16B DMA per lane issue) ---
  for (int i = tid; i < ROWS * 16; i += 256) {       // 2048 float4 tiles
    int r = i >> 4, c = (i & 15) << 2;
    async_copy_b128(&Xs[r * XS + c], base + gbase + (size_t)i * 4);
  }
  // --- async stage: W1 (128x64 -> stride 72), W2 (64x64 -> stride 72) ---
  for (int i = tid; i < 128 * 16; i += 256) {        // 2048 float4 tiles
    int k = i >> 4, c = (i & 15) << 2;
    async_copy_b128(&W1s[k * WS + c], fus_W1 + (size_t)i * 4);
  }
  for (int i = tid; i < 64 * 16; i += 256) {         // 1024 float4 tiles
    int k = i >> 4, c = (i & 15) << 2;
    async_copy_b128(&W2s[k * WS + c], fus_W2 + (size_t)i * 4);
  }

  // --- scalar prologue overlapped with the async DMA ---
  if (tid < 128) gWs[tid] = gate_W[tid];
  if (tid < Hn) { b1s[tid] = fus_b1[tid]; b2s[tid] = fus_b2[tid]; }
  if (tid == 0) {                     // LN stats of the default (agg=0) row
    float mu = 0.0f;
    for (int k = 0; k < Hn; ++k) mu += upd_b[k];
    mu *= (1.0f / Hn);
    float var = 0.0f;
    for (int k = 0; k < Hn; ++k) { float d = upd_b[k] - mu; var += d * d; }
    var *= (1.0f / Hn);
    stats[0] = mu;
    stats[1] = rsqrtf(var + LN_EPS);
  }

  wait_async0();                      // this wave's DMA complete
  __syncthreads();                    // all waves' DMA + prologue visible

  if (tid < Hn)
    dps[tid] = (upd_b[tid] - stats[0]) * stats[1] * ln_g[tid] + ln_b[tid];
  __syncthreads();
  // default prompt row broadcast into X[:,64:128]
  for (int i = tid; i < ROWS * Hn; i += 256) {
    int r = i >> 6, c = i & 63;
    Xs[r * XS + 64 + c] = dps[c];
  }
  __syncthreads();
  // patch the <=20 special rows of this sample that land in this tile
  for (int e = 0; e < En; ++e) {
    int id = special_ids[b * En + e];
    int lr = id - row0;
    if (id >= 0 && lr >= 0 && lr < ROWS && tid < Hn)
      Xs[lr * XS + 64 + tid] = special_rows[(((long)b * En + e) << 6) + tid];
  }
  __syncthreads();
  // gate = sigmoid([base|prompt] . gate_W + gate_b)
  if (tid < ROWS) {
    float g = gate_b[0];
    for (int k = 0; k < 128; ++k) g += Xs[tid * XS + k] * gWs[k];
    gates[tid] = 1.0f / (1.0f + __expf(-g));
  }
  __syncthreads();

  // --- WMMA GEMM 1: X[128x128] @ W1[128x64] ---
  const int lane = tid & 31;
  const int w    = tid >> 5;                 // wave 0..7
  const int m    = lane & 15;                // A row / B col within tile
  const int kh   = (lane >> 4) << 1;         // K sub-offset: 0 or 2
  const int rowA = w * 16 + m;
  const int mhi  = (lane >> 4) << 3;         // +8 rows for upper half-wave (C/D)

  v8f acc[4];
  #pragma unroll
  for (int nt = 0; nt < 4; ++nt)
    #pragma unroll
    for (int q = 0; q < 8; ++q) acc[nt][q] = 0.0f;

  #pragma unroll 4
  for (int k = 0; k < 128; k += 4) {
    v2f a;
    a.x = Xs[rowA * XS + k + kh];
    a.y = Xs[rowA * XS + k + kh + 1];
    #pragma unroll
    for (int nt = 0; nt < 4; ++nt) {
      v2f bb;
      bb.x = W1s[(k + kh) * WS + nt * 16 + m];
      bb.y = W1s[(k + kh + 1) * WS + nt * 16 + m];
      acc[nt] = __builtin_amdgcn_wmma_f32_16x16x4_f32(
          false, a, false, bb, (short)0, acc[nt], false, false);
    }
  }
  __syncthreads();   // everyone done reading X prompt half before Y overlay

  // relu(acc + b1) -> overlay into Xs[:,64:128] (wave-private rows)
  #pragma unroll
  for (int nt = 0; nt < 4; ++nt) {
    #pragma unroll
    for (int q = 0; q < 8; ++q) {
      int M   = q + mhi;
      int col = nt * 16 + m;
      Xs[(w * 16 + M) * XS + 64 + col] = fmaxf(acc[nt][q] + b1s[col], 0.0f);
    }
  }

  // --- WMMA GEMM 2: Y[16x64] @ W2[64x64] (per-wave, LDS in-order) ---
  v8f d[4];
  #pragma unroll
  for (int nt = 0; nt < 4; ++nt)
    #pragma unroll
    for (int q = 0; q < 8; ++q) d[nt][q] = 0.0f;

  #pragma unroll 4
  for (int k = 0; k < 64; k += 4) {
    v2f a;
    a.x = Xs[rowA * XS + 64 + k + kh];
    a.y = Xs[rowA * XS + 64 + k + kh + 1];
    #pragma unroll
    for (int nt = 0; nt < 4; ++nt) {
      v2f bb;
      bb.x = W2s[(k + kh) * WS + nt * 16 + m];
      bb.y = W2s[(k + kh + 1) * WS + nt * 16 + m];
      d[nt] = __builtin_amdgcn_wmma_f32_16x16x4_f32(
          false, a, false, bb, (short)0, d[nt], false, false);
    }
  }

  // --- epilogue: out = g*fused + (1-g)*base ---
  float* outp = out + gbase;
  #pragma unroll
  for (int nt = 0; nt < 4; ++nt) {
    #pragma unroll
    for (int q = 0; q < 8; ++q) {
      int M   = q + mhi;
      int r   = w * 16 + M;
      int col = nt * 16 + m;
      float fused = d[nt][q] + b2s[col];
      float g     = gates[r];
      float bv    = Xs[r * XS + col];          // base, untouched by overlay
      outp[(size_t)r * 64 + col] = g * fused + (1.0f - g) * bv;
    }
  }
}

// ---------------------------------------------------------------------------
extern "C" void kernel_launch(void* const* d_in, const int* in_sizes, int n_in,
                              void* d_out, int out_size, void* d_ws, size_t ws_size,
                              hipStream_t stream) {
  const int*   query = (const int*)  d_in[0];
  const int*   edge  = (const int*)  d_in[1];
  const float* base  = (const float*)d_in[2];
  const float* noise = (const float*)d_in[3];
  const float* msg_W = (const float*)d_in[4];
  const float* msg_b = (const float*)d_in[5];
  const float* upd_W = (const float*)d_in[6];
  const float* upd_b = (const float*)d_in[7];
  const float* ln_g  = (const float*)d_in[8];
  const float* ln_b  = (const float*)d_in[9];
  const float* fW1   = (const float*)d_in[10];
  const float* fb1   = (const float*)d_in[11];
  const float* fW2   = (const float*)d_in[12];
  const float* fb2   = (const float*)d_in[13];
  const float* gW    = (const float*)d_in[14];
  const float* gb    = (const float*)d_in[15];
  float* out = (float*)d_out;

  // workspace layout: [B*E ids][B*E*H special prompt rows]
  int*   special_ids  = (int*)d_ws;
  float* special_rows =
      (float*)((char*)d_ws + ((Bn * En * sizeof(int) + 255) & ~(size_t)255));

  prompt_sparse_kernel<<<Bn, 64, 0, stream>>>(
      query, edge, noise, msg_W, msg_b, upd_W, upd_b, ln_g, ln_b,
      special_ids, special_rows);

  fuse_gate_kernel<<<(Bn * Rn) / ROWS, 256, 0, stream>>>(
      base, fW1, fb1, fW2, fb2, gW, gb, upd_b, ln_g, ln_b,
      special_ids, special_rows, out);
}